// SocialGraph_72730976191047
// MI455X (gfx1250) — compile-verified
//
#include <hip/hip_runtime.h>
#include <hip/hip_bf16.h>

#define NUM_USERS_C 30000
#define NUM_ITEMS_C 30000
#define N_NODES_C   60000
#define N_REL_C     10
#define NUM_BASES_C 8
#define D_C         128
#define E_C         1000000
#define B_C         256
#define L_C         50

typedef __attribute__((ext_vector_type(2))) float v2f;
typedef __attribute__((ext_vector_type(8))) float v8f;

// ---------------------------------------------------------------------------
// D = A(16x4 f32) * B(4x16 f32) + C(16x16 f32), wave32 WMMA
// ---------------------------------------------------------------------------
__device__ __forceinline__ v8f wmma4(v2f a, v2f b, v8f c) {
  return __builtin_amdgcn_wmma_f32_16x16x4_f32(false, a, false, b, (short)0, c,
                                               false, false);
}

// ---------------------------------------------------------------------------
// Pass 1: per-(dst,rel) edge counts + per-item (src) edge counts
// ---------------------------------------------------------------------------
__global__ void count_kernel(const int* __restrict__ esrc,
                             const int* __restrict__ edst,
                             const int* __restrict__ etyp,
                             int* __restrict__ cnt_dr,
                             int* __restrict__ pcnt) {
  int e = blockIdx.x * blockDim.x + threadIdx.x;
  if (e >= E_C) return;
  __builtin_prefetch(esrc + e + 4096, 0, 1);   // speculative global_prefetch_b8
  atomicAdd(&cnt_dr[edst[e] * N_REL_C + etyp[e]], 1);
  atomicAdd(&pcnt[esrc[e] - NUM_USERS_C], 1);
}

// ---------------------------------------------------------------------------
// Pass 2: one wave per edge.  msg = sum_b comp[rel,b]*basis[b,src,:]
//   social[dst,:] += msg / cnt[dst,rel]   (fuses per-rel mean + sum over rel)
// ---------------------------------------------------------------------------
__global__ void msg_kernel(const int* __restrict__ esrc,
                           const int* __restrict__ edst,
                           const int* __restrict__ etyp,
                           const float* __restrict__ basis,
                           const float* __restrict__ comp,
                           const int* __restrict__ cnt_dr,
                           float* __restrict__ social) {
  int e = blockIdx.x * 8 + (threadIdx.x >> 5);
  if (e >= E_C) return;
  int lane = threadIdx.x & 31;
  int src = esrc[e], dst = edst[e], r = etyp[e];
  int c = cnt_dr[dst * N_REL_C + r];
  float inv = 1.0f / (float)(c > 0 ? c : 1);
  int d0 = lane * 4;
  float ax = 0.f, ay = 0.f, az = 0.f, aw = 0.f;
#pragma unroll
  for (int b = 0; b < NUM_BASES_C; ++b) {
    float cb = comp[r * NUM_BASES_C + b];
    const float4 v = *reinterpret_cast<const float4*>(
        basis + ((size_t)b * N_NODES_C + src) * D_C + d0);
    ax += cb * v.x; ay += cb * v.y; az += cb * v.z; aw += cb * v.w;
  }
  float* sp = social + (size_t)dst * D_C + d0;
  atomicAdd(sp + 0, ax * inv);
  atomicAdd(sp + 1, ay * inv);
  atomicAdd(sp + 2, az * inv);
  atomicAdd(sp + 3, aw * inv);
}

// social[u,d] += root[u,d] + bias[d]   (users only)
__global__ void social_finalize_kernel(const float* __restrict__ root,
                                       const float* __restrict__ bias,
                                       float* __restrict__ social) {
  int t = blockIdx.x * blockDim.x + threadIdx.x;
  if (t >= NUM_USERS_C * D_C) return;
  social[t] += root[t] + bias[t & (D_C - 1)];
}

// Pass 3: pooled[src,:] += social[dst,:], one wave per edge
__global__ void pool_kernel(const int* __restrict__ esrc,
                            const int* __restrict__ edst,
                            const float* __restrict__ social,
                            float* __restrict__ pooled) {
  int e = blockIdx.x * 8 + (threadIdx.x >> 5);
  if (e >= E_C) return;
  int lane = threadIdx.x & 31;
  int src = esrc[e] - NUM_USERS_C, dst = edst[e];
  int d0 = lane * 4;
  const float4 v = *reinterpret_cast<const float4*>(
      social + (size_t)dst * D_C + d0);
  float* pp = pooled + (size_t)src * D_C + d0;
  atomicAdd(pp + 0, v.x);
  atomicAdd(pp + 1, v.y);
  atomicAdd(pp + 2, v.z);
  atomicAdd(pp + 3, v.w);
}

__global__ void pooled_finalize_kernel(const int* __restrict__ pcnt,
                                       float* __restrict__ pooled) {
  int t = blockIdx.x * blockDim.x + threadIdx.x;
  if (t >= NUM_ITEMS_C * D_C) return;
  int c = pcnt[t >> 7];
  pooled[t] *= 1.0f / (float)(c > 0 ? c : 1);
}

// h[b,l,:] = valid ? pooled[id] : 0 ; also write valid mask
__global__ void build_h_kernel(const int* __restrict__ ctx,
                               const int* __restrict__ pcnt,
                               const float* __restrict__ pooled,
                               float* __restrict__ h,
                               int* __restrict__ vmask) {
  int t = blockIdx.x * blockDim.x + threadIdx.x;
  if (t >= B_C * L_C * D_C) return;
  int bl = t >> 7, d = t & (D_C - 1);
  int id = ctx[bl];
  int valid = (id >= NUM_USERS_C) && (pcnt[id - NUM_USERS_C] > 0);
  h[t] = valid ? pooled[(size_t)(id - NUM_USERS_C) * D_C + d] : 0.0f;
  if (d == 0) vmask[bl] = valid;
}

// ---------------------------------------------------------------------------
// e[row] = tanh(h @ attn_a) . attn_b    via V_WMMA_F32_16X16X4_F32
// rows = flattened b*L+l (12800).  One wave per 16-row M-tile, all 8 N-tiles.
// ---------------------------------------------------------------------------
__global__ void attn_score_kernel(const float* __restrict__ h,
                                  const float* __restrict__ attn_a,
                                  const float* __restrict__ attn_b,
                                  float* __restrict__ escore) {
  int wave = threadIdx.x >> 5;
  int lane = threadIdx.x & 31;
  int half = lane >> 4, l16 = lane & 15;
  int row0 = (blockIdx.x * 8 + wave) * 16;

  v8f zero = {};
  v8f c[8];
#pragma unroll
  for (int nt = 0; nt < 8; ++nt) c[nt] = zero;

  for (int k0 = 0; k0 < D_C; k0 += 4) {
    int kk = k0 + 2 * half;  // lane's two K values: kk, kk+1
    v2f a = *reinterpret_cast<const v2f*>(h + (size_t)(row0 + l16) * D_C + kk);
#pragma unroll
    for (int nt = 0; nt < 8; ++nt) {
      v2f b;
      b.x = attn_a[(size_t)kk * D_C + nt * 16 + l16];
      b.y = attn_a[(size_t)(kk + 1) * D_C + nt * 16 + l16];
      c[nt] = wmma4(a, b, c[nt]);
    }
  }
  // per-row dot with attn_b: partial per lane, butterfly inside each 16-lane half
  float s[8];
#pragma unroll
  for (int j = 0; j < 8; ++j) s[j] = 0.0f;
#pragma unroll
  for (int nt = 0; nt < 8; ++nt) {
    float bn = attn_b[nt * 16 + l16];
#pragma unroll
    for (int j = 0; j < 8; ++j) s[j] += tanhf(c[nt][j]) * bn;
  }
#pragma unroll
  for (int j = 0; j < 8; ++j) {
    float v = s[j];
    for (int m = 1; m < 16; m <<= 1) v += __shfl_xor(v, m, 32);
    if (l16 == 0) escore[row0 + j + 8 * half] = v;
  }
}

// masked softmax over L=50 + attn-weighted sum -> rep[b,:]
__global__ void softmax_rep_kernel(const float* __restrict__ escore,
                                   const int* __restrict__ vmask,
                                   const float* __restrict__ h,
                                   float* __restrict__ rep) {
  int b = blockIdx.x, t = threadIdx.x;  // 128 threads
  __shared__ float red[128];
  __shared__ float attn_s[L_C];
  float ev = -1e30f;
  int valid = 0;
  if (t < L_C) {
    valid = vmask[b * L_C + t];
    ev = valid ? escore[b * L_C + t] : -1e9f;
  }
  red[t] = ev;
  __syncthreads();
  for (int s = 64; s > 0; s >>= 1) {
    if (t < s) red[t] = fmaxf(red[t], red[t + s]);
    __syncthreads();
  }
  float m = red[0];
  __syncthreads();
  float p = (t < L_C) ? __expf(ev - m) : 0.0f;
  red[t] = p;
  __syncthreads();
  for (int s = 64; s > 0; s >>= 1) {
    if (t < s) red[t] += red[t + s];
    __syncthreads();
  }
  float sum = red[0];
  if (t < L_C) attn_s[t] = valid ? (p / sum) : 0.0f;
  __syncthreads();
  float acc = 0.0f;
  for (int l = 0; l < L_C; ++l)
    acc += attn_s[l] * h[((size_t)b * L_C + l) * D_C + t];
  rep[(size_t)b * D_C + t] = acc;
}

// ---------------------------------------------------------------------------
// out = relu(in @ w^T + bias), in [256,128], w [128,128] — WMMA f32 16x16x4
// ---------------------------------------------------------------------------
__global__ void fc_kernel(const float* __restrict__ in,
                          const float* __restrict__ w,
                          const float* __restrict__ bias,
                          float* __restrict__ out) {
  int wave = threadIdx.x >> 5;
  int lane = threadIdx.x & 31;
  int half = lane >> 4, l16 = lane & 15;
  int row0 = (blockIdx.x * 8 + wave) * 16;

  v8f zero = {};
  v8f c[8];
#pragma unroll
  for (int nt = 0; nt < 8; ++nt) c[nt] = zero;

  for (int k0 = 0; k0 < D_C; k0 += 4) {
    int kk = k0 + 2 * half;
    v2f a = *reinterpret_cast<const v2f*>(in + (size_t)(row0 + l16) * D_C + kk);
#pragma unroll
    for (int nt = 0; nt < 8; ++nt) {
      // B[k][n] = w[n,k]  -> consecutive in k: vectorized v2f load
      v2f b = *reinterpret_cast<const v2f*>(
          w + (size_t)(nt * 16 + l16) * D_C + kk);
      c[nt] = wmma4(a, b, c[nt]);
    }
  }
#pragma unroll
  for (int nt = 0; nt < 8; ++nt) {
    int n = nt * 16 + l16;
    float bn = bias[n];
#pragma unroll
    for (int j = 0; j < 8; ++j) {
      int row = row0 + j + 8 * half;
      float v = c[nt][j] + bn;
      out[(size_t)row * D_C + n] = v > 0.0f ? v : 0.0f;
    }
  }
}

// ---------------------------------------------------------------------------
extern "C" void kernel_launch(void* const* d_in, const int* in_sizes, int n_in,
                              void* d_out, int out_size, void* d_ws,
                              size_t ws_size, hipStream_t stream) {
  const int*   ctx    = (const int*)d_in[0];
  const int*   esrc   = (const int*)d_in[1];
  const int*   edst   = (const int*)d_in[2];
  const int*   etyp   = (const int*)d_in[3];
  const float* basis  = (const float*)d_in[4];
  const float* comp   = (const float*)d_in[5];
  const float* root   = (const float*)d_in[6];
  const float* bias   = (const float*)d_in[7];
  const float* attn_a = (const float*)d_in[8];
  const float* attn_b = (const float*)d_in[9];
  const float* fc1_w  = (const float*)d_in[10];
  const float* fc1_b  = (const float*)d_in[11];
  const float* fc2_w  = (const float*)d_in[12];
  const float* fc2_b  = (const float*)d_in[13];

  float* proj = (float*)d_out;                 // [256,128]
  float* h    = (float*)d_out + B_C * D_C;     // [256,50,128]

  char* ws = (char*)d_ws;
  size_t off = 0;
  auto take = [&](size_t bytes) -> void* {
    void* p = ws + off;
    off = (off + bytes + 255) & ~(size_t)255;
    return p;
  };
  float* social = (float*)take(sizeof(float) * NUM_USERS_C * D_C);
  float* pooled = (float*)take(sizeof(float) * NUM_ITEMS_C * D_C);
  int*   cnt_dr = (int*)take(sizeof(int) * NUM_USERS_C * N_REL_C);
  int*   pcnt   = (int*)take(sizeof(int) * NUM_ITEMS_C);
  float* escore = (float*)take(sizeof(float) * B_C * L_C);
  int*   vmask  = (int*)take(sizeof(int) * B_C * L_C);
  float* rep    = (float*)take(sizeof(float) * B_C * D_C);
  float* xbuf   = (float*)take(sizeof(float) * B_C * D_C);

  hipMemsetAsync(social, 0, sizeof(float) * NUM_USERS_C * D_C, stream);
  hipMemsetAsync(pooled, 0, sizeof(float) * NUM_ITEMS_C * D_C, stream);
  hipMemsetAsync(cnt_dr, 0, sizeof(int) * NUM_USERS_C * N_REL_C, stream);
  hipMemsetAsync(pcnt, 0, sizeof(int) * NUM_ITEMS_C, stream);

  count_kernel<<<(E_C + 255) / 256, 256, 0, stream>>>(esrc, edst, etyp, cnt_dr,
                                                      pcnt);
  msg_kernel<<<(E_C + 7) / 8, 256, 0, stream>>>(esrc, edst, etyp, basis, comp,
                                                cnt_dr, social);
  social_finalize_kernel<<<(NUM_USERS_C * D_C + 255) / 256, 256, 0, stream>>>(
      root, bias, social);
  pool_kernel<<<(E_C + 7) / 8, 256, 0, stream>>>(esrc, edst, social, pooled);
  pooled_finalize_kernel<<<(NUM_ITEMS_C * D_C + 255) / 256, 256, 0, stream>>>(
      pcnt, pooled);
  build_h_kernel<<<(B_C * L_C * D_C + 255) / 256, 256, 0, stream>>>(
      ctx, pcnt, pooled, h, vmask);
  // 12800 rows -> 800 M-tiles -> 100 blocks of 8 waves
  attn_score_kernel<<<(B_C * L_C) / 128, 256, 0, stream>>>(h, attn_a, attn_b,
                                                           escore);
  softmax_rep_kernel<<<B_C, 128, 0, stream>>>(escore, vmask, h, rep);
  // 256 rows -> 16 M-tiles -> 2 blocks of 8 waves
  fc_kernel<<<2, 256, 0, stream>>>(rep, fc1_w, fc1_b, xbuf);
  fc_kernel<<<2, 256, 0, stream>>>(xbuf, fc2_w, fc2_b, proj);
}